// LFQ_1726576854532
// MI455X (gfx1250) — compile-verified
//
#include <hip/hip_runtime.h>

// ---------------------------------------------------------------------------
// LFQ forward on gfx1250 (MI455X):
//   k1: h = x@W_in + b_in   -- v_wmma_f32_16x16x4_f32, A staged via
//                              global_load_async_to_lds_b128 double buffer
//   k2: factorized softmax entropy (Gray-code enumeration of 2^14 codes)
//   k3: bit-pack indices
//   k4: out = sign(h)@W_out + b_out via v_wmma_f32_16x16x4_f32
//   k5: finalize codebook entropy + aux loss
// ---------------------------------------------------------------------------

typedef float v2f __attribute__((ext_vector_type(2)));
typedef float v8f __attribute__((ext_vector_type(8)));
typedef int   v4i __attribute__((vector_size(16)));          // b128 payload
typedef __attribute__((address_space(1))) v4i* gv4i_p;       // global AS
typedef __attribute__((address_space(3))) v4i* lv4i_p;       // LDS AS

#define M_TOTAL 16384   // B*N
#define DIM     512
#define CD      14      // codebook dim
#define CPAD    16      // padded codebook dim for WMMA
#define JSIZE   16384   // 2^14 codes
#define LOG_EPS_LOG (-11.512925464970229f)   // log(1e-5)

#define KCHUNK  64      // K tile staged in LDS per pipeline stage
#define ASTR    68      // sA row pitch (floats): banks 4r..4r+3 -> 64 distinct
#define WSTR    516     // sWt row pitch (floats): banks 4c..4c+3 -> 64 distinct

#if defined(__has_builtin)
#  if __has_builtin(__builtin_amdgcn_global_load_async_to_lds_b128) && \
      __has_builtin(__builtin_amdgcn_s_wait_asynccnt)
#    define HAVE_ASYNC_LDS 1
#  else
#    define HAVE_ASYNC_LDS 0
#  endif
#else
#  define HAVE_ASYNC_LDS 0
#endif

// ------------------------- GEMM in: h = x @ W_in + b_in ---------------------
__global__ __launch_bounds__(256) void lfq_gemm_in(
    const float* __restrict__ x, const float* __restrict__ Win,
    const float* __restrict__ bin, float* __restrict__ h, float* __restrict__ hq)
{
  __shared__ float sWt[CPAD * WSTR];             // W_in^T, N-padded: [n][k]
  __shared__ float sA[2][8][16 * ASTR];          // per-wave A double buffer

  const int tid = threadIdx.x;
  for (int i = tid; i < CPAD * DIM; i += 256) {  // transpose + zero-pad W_in
    const int n = i >> 9, k = i & (DIM - 1);
    sWt[n * WSTR + k] = (n < CD) ? Win[k * CD + n] : 0.f;
  }

  const int lane  = tid & 31;
  const int wave  = tid >> 5;
  const int m0    = (blockIdx.x * 8 + wave) * 16;
  const int row   = lane & 15;                   // A/C: M within tile
  const int khalf = (lane >> 4) * 2;             // A/B: K sub-offset 0 or 2
  const int col   = lane & 15;                   // B/C: N within tile

  float* const sA0 = &sA[0][wave][0];
  float* const sA1 = &sA[1][wave][0];

  // stage one 16-row x 64-col f32 chunk of A (wave-local): 8 x b128 per lane
  auto issue = [&](float* dstbuf, int kc) {
#pragma unroll
    for (int i = 0; i < 8; ++i) {
      const int id = lane + 32 * i;              // 0..255
      const int r  = id >> 4;                    // 0..15
      const int c4 = (id & 15) << 2;             // 0,4,...,60
      const float* g = x + (size_t)(m0 + r) * DIM + kc + c4;
      float*       l = dstbuf + r * ASTR + c4;
#if HAVE_ASYNC_LDS
      __builtin_amdgcn_global_load_async_to_lds_b128((gv4i_p)g, (lv4i_p)l, 0, 0);
#else
      const float4 v = *(const float4*)g;
      *(float4*)l = v;
#endif
    }
  };

  __syncthreads();                               // sWt ready

  issue(sA0, 0);
  v8f c = {};
  for (int ch = 0; ch < DIM / KCHUNK; ++ch) {
    float* const aw = (ch & 1) ? sA1 : sA0;
    if (ch + 1 < DIM / KCHUNK) {
      issue((ch & 1) ? sA0 : sA1, (ch + 1) * KCHUNK);
#if HAVE_ASYNC_LDS
      __builtin_amdgcn_s_wait_asynccnt(8);       // previous chunk complete
#endif
    } else {
#if HAVE_ASYNC_LDS
      __builtin_amdgcn_s_wait_asynccnt(0);       // last chunk complete
#endif
    }
#pragma unroll
    for (int k = 0; k < KCHUNK; k += 4) {
      const v2f a = *(const v2f*)&aw[row * ASTR + k + khalf];
      const v2f b = *(const v2f*)&sWt[col * WSTR + ch * KCHUNK + k + khalf];
      c = __builtin_amdgcn_wmma_f32_16x16x4_f32(false, a, false, b, (short)0, c,
                                                false, false);
    }
  }

  const float bb = (col < CD) ? bin[col] : 0.f;
#pragma unroll
  for (int r = 0; r < 8; ++r) {
    const int m = m0 + r + ((lane >> 4) << 3);
    const float hv = c[r] + bb;                  // padded cols are exactly 0
    h [(size_t)m * CPAD + col] = hv;
    hq[(size_t)m * CPAD + col] = (col < CD) ? (hv > 0.f ? 1.f : -1.f) : 0.f;
  }
}

// --------------------- entropy via factorized softmax -----------------------
// p_j = prod_d q_d(bit_d(j));  log p_j updated by one add per j (Gray code).
__global__ __launch_bounds__(256) void lfq_entropy(
    const float* __restrict__ h, float* __restrict__ g_avg,
    float* __restrict__ g_ps)
{
  __shared__ float sAcc[JSIZE];                  // 64 KB avg_prob partial
  __shared__ float sDiff[8][16];                 // per-wave lp1-lp0, bits 5..13
  const int tid = threadIdx.x;
  for (int i = tid; i < JSIZE; i += 256) sAcc[i] = 0.f;
  __syncthreads();

  const int lane = tid & 31;
  const int wave = tid >> 5;
  float acc = 0.f;                               // sum p * clamped_log(p)

  for (int s = wave; s < 256; s += 8) {          // one wave per sample
    const int i = blockIdx.x * 256 + s;
    const float* hrow = h + (size_t)i * CPAD;
    float lp0[CD], lp1[CD];                      // log q(bit=0/1), bit pos b
#pragma unroll
    for (int d = 0; d < CD; ++d) {
      const int b = CD - 1 - d;                  // component d <-> bit 13-d
      const float z = 400.f * hrow[d];
      // stable: softplus(z); softplus(-z) = softplus(z) - z
      const float sp = (z > 0.f) ? (z + log1pf(__expf(-z))) : log1pf(__expf(z));
      lp1[b] = -(sp - z);                        // log sigmoid(z)
      lp0[b] = -sp;                              // log sigmoid(-z)
    }
    float base = 0.f;                            // low 5 bits == lane bits
#pragma unroll
    for (int b = 0; b < 5; ++b) base += ((lane >> b) & 1) ? lp1[b] : lp0[b];
    float hi = 0.f;                              // bits 5..13, Gray state 0
#pragma unroll
    for (int b = 5; b < CD; ++b) hi += lp0[b];
    if (lane == 0) {
#pragma unroll
      for (int b = 5; b < CD; ++b) sDiff[wave][b - 5] = lp1[b] - lp0[b];
    }

    unsigned gray = 0;
    {                                            // t = 0
      const float logp = base + hi;
      const float p = __expf(logp);
      acc += p * fmaxf(logp, LOG_EPS_LOG);
      if (p != 0.f) atomicAdd(&sAcc[lane], p);
    }
    for (unsigned t = 1; t < 512; ++t) {
      const int b = __builtin_ctz(t);
      gray ^= (1u << b);
      const float df = sDiff[wave][b];
      hi += ((gray >> b) & 1u) ? df : -df;
      const int j = (int)(gray << 5) | lane;     // 32 distinct LDS banks
      const float logp = base + hi;
      const float p = __expf(logp);
      acc += p * fmaxf(logp, LOG_EPS_LOG);
      if (p != 0.f) atomicAdd(&sAcc[j], p);      // near-one-hot: rarely taken
    }
  }
  __syncthreads();
  for (int i = tid; i < JSIZE; i += 256) atomicAdd(&g_avg[i], sAcc[i]);
  atomicAdd(g_ps, -acc);                         // entropy = -sum p*logclamp
}

// ------------------------------ indices -------------------------------------
__global__ void lfq_indices(const float* __restrict__ h,
                            float* __restrict__ outIdx)
{
  const int row = blockIdx.x * blockDim.x + threadIdx.x;
  if (row < M_TOTAL) {
    int idx = 0;
#pragma unroll
    for (int d = 0; d < CD; ++d)
      idx |= (h[(size_t)row * CPAD + d] > 0.f) ? (1 << (CD - 1 - d)) : 0;
    outIdx[row] = (float)idx;                    // exact (< 2^14)
  }
}

// -------------------- GEMM out: out = hq @ W_out + b_out --------------------
__global__ __launch_bounds__(256) void lfq_gemm_out(
    const float* __restrict__ hq, const float* __restrict__ Wout,
    const float* __restrict__ bout, float* __restrict__ out)
{
  __shared__ float sW[CPAD * DIM];               // 32 KB, K padded to 16
  const int tid = threadIdx.x;
  for (int i = tid; i < CPAD * DIM; i += 256) {
    const int k = i >> 9;                        // i = k*512 + n
    sW[i] = (k < CD) ? Wout[i] : 0.f;
  }
  __syncthreads();

  const int lane  = tid & 31;
  const int wave  = tid >> 5;
  const int m0    = blockIdx.x * 16;
  const int n0    = (blockIdx.y * 8 + wave) * 16;
  const int row   = lane & 15;
  const int khalf = (lane >> 4) * 2;
  const int col   = lane & 15;

  v8f c = {};
  const float* arow = hq + (size_t)(m0 + row) * CPAD;
#pragma unroll
  for (int k = 0; k < CPAD; k += 4) {
    const v2f a = *(const v2f*)&arow[k + khalf];
    v2f b;
    b.x = sW[(k + khalf)     * DIM + n0 + col];
    b.y = sW[(k + khalf + 1) * DIM + n0 + col];
    c = __builtin_amdgcn_wmma_f32_16x16x4_f32(false, a, false, b, (short)0, c,
                                              false, false);
  }
  const float bb = bout[n0 + col];
#pragma unroll
  for (int r = 0; r < 8; ++r) {
    const int m = m0 + r + ((lane >> 4) << 3);
    out[(size_t)m * DIM + n0 + col] = c[r] + bb;
  }
}

// ------------------------------ finalize ------------------------------------
__global__ __launch_bounds__(256) void lfq_final(
    const float* __restrict__ g_avg, const float* __restrict__ g_ps,
    float* __restrict__ outAux)
{
  __shared__ float red[256];
  const int tid = threadIdx.x;
  const float invM = 1.f / (float)M_TOTAL;
  float ce = 0.f;
  for (int j = tid; j < JSIZE; j += 256) {
    const float a = g_avg[j] * invM;
    ce += -a * logf(fmaxf(a, 1e-5f));
  }
  red[tid] = ce;
  __syncthreads();
  for (int s = 128; s > 0; s >>= 1) {
    if (tid < s) red[tid] += red[tid + s];
    __syncthreads();
  }
  if (tid == 0) {
    const float ps = g_ps[0] * invM;
    outAux[0] = 0.1f * (ps - red[0]);            // ENTROPY_LOSS_WEIGHT=0.1
  }
}

// ------------------------------ init ----------------------------------------
__global__ void lfq_zero(float* __restrict__ p, int n)
{
  const int i = blockIdx.x * blockDim.x + threadIdx.x;
  if (i < n) p[i] = 0.f;
}

// ---------------------------------------------------------------------------
extern "C" void kernel_launch(void* const* d_in, const int* in_sizes, int n_in,
                              void* d_out, int out_size, void* d_ws,
                              size_t ws_size, hipStream_t stream)
{
  const float* x    = (const float*)d_in[0];   // [8,2048,512]
  const float* Win  = (const float*)d_in[1];   // [512,14]
  const float* bin  = (const float*)d_in[2];   // [14]
  const float* Wout = (const float*)d_in[3];   // [14,512]
  const float* bout = (const float*)d_in[4];   // [512]

  float* out    = (float*)d_out;                           // 8*2048*512
  float* outIdx = out + (size_t)M_TOTAL * DIM;             // 16384
  float* outAux = outIdx + M_TOTAL;                        // 1

  float* h     = (float*)d_ws;                             // 16384*16
  float* hq    = h + (size_t)M_TOTAL * CPAD;               // 16384*16
  float* g_avg = hq + (size_t)M_TOTAL * CPAD;              // 16384
  float* g_ps  = g_avg + JSIZE;                            // 1 (+pad)

  lfq_zero<<<(JSIZE + 4 + 255) / 256, 256, 0, stream>>>(g_avg, JSIZE + 4);
  lfq_gemm_in<<<128, 256, 0, stream>>>(x, Win, bin, h, hq);
  lfq_entropy<<<64, 256, 0, stream>>>(h, g_avg, g_ps);
  lfq_indices<<<M_TOTAL / 256, 256, 0, stream>>>(h, outIdx);
  lfq_gemm_out<<<dim3(1024, 4), 256, 0, stream>>>(hq, Wout, bout, out);
  lfq_final<<<1, 256, 0, stream>>>(g_avg, g_ps, outAux);
}